// GraphEncoder_46076409151867
// MI455X (gfx1250) — compile-verified
//
#include <hip/hip_runtime.h>
#include <hip/hip_bf16.h>

#define NGENES 5000
#define NBATCH 32
#define HFEAT  64

typedef float v2f __attribute__((ext_vector_type(2)));
typedef float v8f __attribute__((ext_vector_type(8)));

__device__ __forceinline__ int clamp_idx(long long v) {
    if (v < 0) v = 0;
    if (v >= NGENES) v = NGENES - 1;
    return (int)v;
}

// ---------------- zero the (small) accumulators: cnt, cursor, pooled ----------------
__global__ void k_zero32(unsigned int* __restrict__ p, int n) {
    int i = blockIdx.x * blockDim.x + threadIdx.x;
    int stride = gridDim.x * blockDim.x;
    for (; i < n; i += stride) p[i] = 0u;
}

// ---------------- transpose x [B,N] -> xT [N,B] ----------------
__global__ void k_xT(const float* __restrict__ x, float* __restrict__ xT) {
    int i = blockIdx.x * blockDim.x + threadIdx.x;
    if (i >= NGENES * NBATCH) return;
    int n = i >> 5, b = i & 31;
    xT[i] = x[b * NGENES + n];
}

// ---------------- integer degree counts over dst (edges + self loops) ----------------
__global__ void k_count(const long long* __restrict__ ei, int E, int* __restrict__ cnt) {
    int e = blockIdx.x * blockDim.x + threadIdx.x;
    if (e >= E + NGENES) return;
    int dst = (e < E) ? clamp_idx(ei[E + e]) : (e - E);
    atomicAdd(&cnt[dst], 1);
}

// ---------------- single-block exclusive scan: rowptr[0..NGENES] ----------------
__global__ void k_scan(const int* __restrict__ cnt, int* __restrict__ rowptr) {
    __shared__ int part[256];
    const int tid = threadIdx.x;
    const int CH = (NGENES + 255) / 256;  // 20
    int base = tid * CH;
    int sum = 0;
    for (int i = 0; i < CH; ++i) {
        int idx = base + i;
        sum += (idx < NGENES) ? cnt[idx] : 0;
    }
    part[tid] = sum;
    __syncthreads();
    if (tid == 0) {
        int acc = 0;
        for (int i = 0; i < 256; ++i) { int t = part[i]; part[i] = acc; acc += t; }
        rowptr[NGENES] = acc;
    }
    __syncthreads();
    int off = part[tid];
    for (int i = 0; i < CH; ++i) {
        int idx = base + i;
        if (idx < NGENES) { rowptr[idx] = off; off += cnt[idx]; }
    }
}

// ---------------- dinv[n] = 1/sqrt(deg) ----------------
__global__ void k_dinv(const int* __restrict__ cnt, float* __restrict__ dinv) {
    int n = blockIdx.x * blockDim.x + threadIdx.x;
    if (n >= NGENES) return;
    float d = (float)cnt[n];
    dinv[n] = (d > 0.0f) ? (1.0f / sqrtf(d)) : 0.0f;
}

// ---------------- fill CSR records {src, norm} grouped by dst ----------------
__global__ void k_fill(const long long* __restrict__ ei, int E,
                       const int* __restrict__ rowptr, int* __restrict__ cursor,
                       const float* __restrict__ dinv,
                       int* __restrict__ csr_src, float* __restrict__ csr_nrm) {
    int e = blockIdx.x * blockDim.x + threadIdx.x;
    if (e >= E + NGENES) return;
    int src, dst;
    if (e < E) { src = clamp_idx(ei[e]); dst = clamp_idx(ei[E + e]); }
    else       { src = dst = e - E; }
    int pos = rowptr[dst] + atomicAdd(&cursor[dst], 1);
    csr_src[pos] = src;
    csr_nrm[pos] = dinv[src] * dinv[dst];
}

// ---------- layer-1 aggregation (gather, no fp atomics) ----------
// s[dst,b] = sum over incoming edges of norm * x[b, src]
__global__ void k_l1(const int* __restrict__ rowptr,
                     const int* __restrict__ csr_src, const float* __restrict__ csr_nrm,
                     const float* __restrict__ xT, float* __restrict__ s) {
    int t = blockIdx.x * blockDim.x + threadIdx.x;
    if (t >= NGENES * NBATCH) return;
    int dst = t >> 5;           // uniform per wave
    int b   = t & 31;
    int beg = rowptr[dst], end = rowptr[dst + 1];
    float acc = 0.0f;
    for (int i = beg; i < end; ++i)
        acc += csr_nrm[i] * xT[csr_src[i] * NBATCH + b];
    s[t] = acc;
}

// ---------- layer-2 aggregation of on-the-fly h1 (gather, no fp atomics) ----------
// h1[src,b,f] = relu(s[src,b]*W1[f] + b1[f]);  a1[dst,b,f] = sum norm * h1[src,b,f]
__global__ void k_l2(const int* __restrict__ rowptr,
                     const int* __restrict__ csr_src, const float* __restrict__ csr_nrm,
                     const float* __restrict__ s,
                     const float* __restrict__ W1, const float* __restrict__ b1,
                     float* __restrict__ a1) {
    int t = blockIdx.x * blockDim.x + threadIdx.x;
    if (t >= NGENES * 512) return;
    int dst = t >> 9;           // uniform per wave (512 threads per dst)
    int r   = t & 511;
    int b   = r >> 4;           // 0..31
    int f4  = (r & 15) << 2;    // 0,4,...,60
    float w0 = W1[f4 + 0], w1 = W1[f4 + 1], w2 = W1[f4 + 2], w3 = W1[f4 + 3];
    float c0 = b1[f4 + 0], c1 = b1[f4 + 1], c2 = b1[f4 + 2], c3 = b1[f4 + 3];
    int beg = rowptr[dst], end = rowptr[dst + 1];
    float a0 = 0.f, a1v = 0.f, a2 = 0.f, a3 = 0.f;
    for (int i = beg; i < end; ++i) {
        float norm = csr_nrm[i];
        float sv = s[csr_src[i] * NBATCH + b];
        float h0 = fmaf(sv, w0, c0); h0 = h0 > 0.f ? h0 : 0.f;
        float h1 = fmaf(sv, w1, c1); h1 = h1 > 0.f ? h1 : 0.f;
        float h2 = fmaf(sv, w2, c2); h2 = h2 > 0.f ? h2 : 0.f;
        float h3 = fmaf(sv, w3, c3); h3 = h3 > 0.f ? h3 : 0.f;
        a0 = fmaf(norm, h0, a0); a1v = fmaf(norm, h1, a1v);
        a2 = fmaf(norm, h2, a2); a3  = fmaf(norm, h3, a3);
    }
    float4 out; out.x = a0; out.y = a1v; out.z = a2; out.w = a3;
    *reinterpret_cast<float4*>(a1 + ((long long)dst * NBATCH + b) * HFEAT + f4) = out;
}

// ---------- h2 = relu(a1 @ W2 + b2) via V_WMMA_F32_16X16X4_F32 ----------
// a1: [160000, 64] row-major (row = node*32 + batch); writes gene_embeddings
// directly into d_out in [B, N, F] order.
__global__ void __launch_bounds__(256) k_wmma(const float* __restrict__ a1,
                       const float* __restrict__ W2, const float* __restrict__ b2,
                       float* __restrict__ ge) {
    const int wave = (int)((blockIdx.x * blockDim.x + threadIdx.x) >> 5);
    const int lane = threadIdx.x & 31;
    const int rowTile = wave >> 2;          // 0..9999
    const int c0 = (wave & 3) * 16;         // output column tile base
    if (rowTile >= (NGENES * NBATCH) / 16) return;  // uniform per wave

    const int r15   = lane & 15;
    const int khalf = lane >> 4;            // 0: K pair {0,1}; 1: K pair {2,3}
    const int row   = rowTile * 16 + r15;
    const int ncol  = c0 + r15;

    v8f c = {0.f, 0.f, 0.f, 0.f, 0.f, 0.f, 0.f, 0.f};

#if __has_builtin(__builtin_amdgcn_wmma_f32_16x16x4_f32)
    // A fragment: lane holds A[row][4k + 2*khalf + {0,1}]
    const float2* arow =
        reinterpret_cast<const float2*>(a1 + (long long)row * HFEAT) + khalf;
#pragma unroll
    for (int k = 0; k < 16; ++k) {
        float2 af = arow[2 * k];
        v2f a; a.x = af.x; a.y = af.y;
        v2f b;
        b.x = W2[(4 * k + 2 * khalf + 0) * HFEAT + ncol];
        b.y = W2[(4 * k + 2 * khalf + 1) * HFEAT + ncol];
        c = __builtin_amdgcn_wmma_f32_16x16x4_f32(
                false, a, false, b, (short)0, c, false, false);
    }
#else
    // Scalar fallback with identical D layout: D(m=v+8*khalf, n=ncol)
#pragma unroll
    for (int v = 0; v < 8; ++v) {
        int m = v + 8 * khalf;
        const float* ar = a1 + ((long long)rowTile * 16 + m) * HFEAT;
        float acc = 0.0f;
        for (int k = 0; k < HFEAT; ++k) acc += ar[k] * W2[k * HFEAT + ncol];
        c[v] = acc;
    }
#endif

    const float bv = b2[ncol];
#pragma unroll
    for (int v = 0; v < 8; ++v) {
        int m = v + 8 * khalf;               // D-matrix VGPR layout
        int R = rowTile * 16 + m;            // global row = node*32 + batch
        int node = R >> 5, bb = R & 31;
        float val = c[v] + bv;
        val = val > 0.0f ? val : 0.0f;
        ge[(long long)bb * (NGENES * HFEAT) + node * HFEAT + ncol] = val;
    }
}

// -------- pooled[b,f] (sum form): chunked partial sums over nodes --------
#define PCHUNK 100
__global__ void k_pool(const float* __restrict__ ge, float* __restrict__ pooled) {
    int t = blockIdx.x * blockDim.x + threadIdx.x;
    const int NCHUNK = NGENES / PCHUNK;     // 50
    if (t >= NBATCH * HFEAT * NCHUNK) return;
    int c = t >> 11;                        // chunk id
    int j = t & 2047;                       // (b,f)
    int b = j >> 6, f = j & 63;
    const float* p = ge + (long long)b * NGENES * HFEAT + (long long)c * PCHUNK * HFEAT + f;
    float acc = 0.0f;
    for (int n = 0; n < PCHUNK; ++n) acc += p[n * HFEAT];
    atomicAdd(&pooled[j], acc);
}

// ---------------- mu / logvar heads (1/N scale folded in) ----------------
__global__ void k_head(const float* __restrict__ pooled,
                       const float* __restrict__ Wmu, const float* __restrict__ bmu,
                       const float* __restrict__ Wlv, const float* __restrict__ blv,
                       float* __restrict__ out) {
    int t = blockIdx.x * blockDim.x + threadIdx.x;
    if (t >= 2 * NBATCH * HFEAT) return;
    int o = t >> 11;            // 0 = mu, 1 = logvar
    int j = t & 2047;
    int b = j >> 6, f = j & 63;
    const float* W  = o ? Wlv : Wmu;
    const float* bs = o ? blv : bmu;
    const float inv_n = 1.0f / NGENES;
    float acc = bs[f];
    const float* pr = pooled + b * HFEAT;
    for (int k = 0; k < HFEAT; ++k) acc += (pr[k] * inv_n) * W[k * HFEAT + f];
    out[t] = acc;               // mu at [0,2048), logvar at [2048,4096)
}

extern "C" void kernel_launch(void* const* d_in, const int* in_sizes, int n_in,
                              void* d_out, int out_size, void* d_ws, size_t ws_size,
                              hipStream_t stream) {
    const float*     x   = (const float*)d_in[0];
    const long long* ei  = (const long long*)d_in[1];   // int64 per reference
    const float*     W1  = (const float*)d_in[2];
    const float*     b1  = (const float*)d_in[3];
    const float*     W2  = (const float*)d_in[4];
    const float*     b2  = (const float*)d_in[5];
    const float*     Wmu = (const float*)d_in[6];
    const float*     bmu = (const float*)d_in[7];
    const float*     Wlv = (const float*)d_in[8];
    const float*     blv = (const float*)d_in[9];
    const int E = in_sizes[1] / 2;          // 80000
    const int ET = E + NGENES;              // edges + self loops

    // workspace layout (4-byte units)
    float* ws      = (float*)d_ws;
    int*   cnt     = (int*)ws;              // 5000
    int*   rowptr  = (int*)ws + 5056;       // 5001
    int*   cursor  = (int*)ws + 10112;      // 5000
    float* dinv    = ws + 15168;            // 5000
    float* xT      = ws + 20224;            // 160000
    float* s       = ws + 180224;           // 160000
    int*   csr_src = (int*)ws + 340224;     // 85008
    float* csr_nrm = ws + 425232;           // 85008
    float* pooled  = ws + 510240;           // 2048
    float* a1      = ws + 512288;           // 10,240,000 (16B aligned)

    float* out = (float*)d_out;
    float* ge  = out + 2 * NBATCH * HFEAT;  // gene_embeddings at offset 4096

    // 1) zero cnt + cursor + pooled (harness does not re-poison between replays)
    k_zero32<<<48, 256, 0, stream>>>((unsigned int*)ws, 10112 + 5000);
    k_zero32<<<8, 256, 0, stream>>>((unsigned int*)pooled, NBATCH * HFEAT);

    // 2) transpose x -> xT[n,b]
    k_xT<<<(NGENES * NBATCH + 255) / 256, 256, 0, stream>>>(x, xT);

    // 3) CSR build: count -> scan -> dinv -> fill
    k_count<<<(ET + 255) / 256, 256, 0, stream>>>(ei, E, cnt);
    k_scan<<<1, 256, 0, stream>>>(cnt, rowptr);
    k_dinv<<<(NGENES + 255) / 256, 256, 0, stream>>>(cnt, dinv);
    k_fill<<<(ET + 255) / 256, 256, 0, stream>>>(ei, E, rowptr, cursor, dinv,
                                                 csr_src, csr_nrm);

    // 4) layer-1 gather (scalar per (dst,b))
    k_l1<<<(NGENES * NBATCH + 255) / 256, 256, 0, stream>>>(rowptr, csr_src,
                                                            csr_nrm, xT, s);

    // 5) layer-2 gather of on-the-fly h1 (512 threads per dst)
    k_l2<<<(NGENES * 512) / 256, 256, 0, stream>>>(rowptr, csr_src, csr_nrm,
                                                   s, W1, b1, a1);

    // 6) h2 = relu(a1 @ W2 + b2) via fp32 WMMA; 40000 waves exactly
    k_wmma<<<5000, 256, 0, stream>>>(a1, W2, b2, ge);

    // 7) chunked mean-pool over nodes (sum; scale folded into heads)
    k_pool<<<(NBATCH * HFEAT * (NGENES / PCHUNK) + 255) / 256, 256, 0, stream>>>(
        ge, pooled);

    // 8) mu / logvar heads
    k_head<<<(2 * NBATCH * HFEAT + 255) / 256, 256, 0, stream>>>(
        pooled, Wmu, bmu, Wlv, blv, out);
}